// RandAugment_dct_49237505082026
// MI455X (gfx1250) — compile-verified
//
#include <hip/hip_runtime.h>

// RandAugment DCT brightness/contrast — MI455X (gfx1250, wave32).
// Bandwidth-bound: ~201 MB traffic => ~8.6 us at 23.3 TB/s. Stream with NT
// b128 loads/stores (footprint > 192 MB L2). DC-mean reduction accumulated
// through V_WMMA_F32_16X16X4_F32 (A=ones => layout-independent column sums).

typedef __attribute__((ext_vector_type(2))) float v2f;
typedef __attribute__((ext_vector_type(4))) float v4f;
typedef __attribute__((ext_vector_type(8))) float v8f;

#define CMIN -1024.0f
#define CMAX 1016.0f

__device__ __forceinline__ float clampf(float x) {
    return fminf(fmaxf(x, CMIN), CMAX);
}

__device__ __forceinline__ v4f clamp4(v4f x) {
    x.x = clampf(x.x);
    x.y = clampf(x.y);
    x.z = clampf(x.z);
    x.w = clampf(x.w);
    return x;
}

// ---------------------------------------------------------------------------
// Kernel 1: per-block partial sums of clamped chroma DC coefficients.
// Each wave feeds 64 clamped DC values per iteration into a WMMA B operand;
// with A = ones, D[m][n] = sum_k B[k][n] + C[m][n]. Summing acc[0] over all
// 32 lanes counts rows M=0 and M=8 (identical), hence the 0.5 factor — the
// total is independent of the exact B VGPR layout.
// grid = (groupsPerChannel, 2), block = 256 (8 waves)
// ---------------------------------------------------------------------------
__global__ void dc_partial_kernel(const float* __restrict__ c,
                                  float* __restrict__ partial,
                                  int dcPerBlock, long long chanElems) {
    __shared__ float smem[8];
    const int lane   = threadIdx.x & 31;
    const int wave   = threadIdx.x >> 5;
    const int nWaves = blockDim.x >> 5;

    const long long chOff  = (long long)blockIdx.y * chanElems;
    const int perWave = dcPerBlock / nWaves;   // DC values handled by this wave
    const int iters   = perWave >> 6;          // 64 values per WMMA
    const int baseDC  = blockIdx.x * dcPerBlock + wave * perWave;

    v2f ones = {1.0f, 1.0f};
    v8f acc  = {0.f, 0.f, 0.f, 0.f, 0.f, 0.f, 0.f, 0.f};

    for (int it = 0; it < iters; ++it) {
        const int d = baseDC + (it << 6) + (lane << 1);
        // DC coefficient of every 8x8 block lives at stride 64 floats
        float x0 = clampf(__builtin_nontemporal_load(c + chOff + (long long)d * 64));
        float x1 = clampf(__builtin_nontemporal_load(c + chOff + (long long)(d + 1) * 64));
        v2f b = {x0, x1};
        // (neg_a, A, neg_b, B, c_mod, C, reuse_a, reuse_b)
        acc = __builtin_amdgcn_wmma_f32_16x16x4_f32(
            false, ones, false, b, (short)0, acc, false, false);
    }

    float v = acc[0] * 0.5f;                    // undo M=0 / M=8 double count
    for (int off = 16; off > 0; off >>= 1)      // wave32 butterfly
        v += __shfl_xor(v, off, 32);

    if (lane == 0) smem[wave] = v;
    __syncthreads();
    if (threadIdx.x == 0) {
        float s = 0.0f;
        for (int w = 0; w < nWaves; ++w) s += smem[w];
        partial[blockIdx.y * gridDim.x + blockIdx.x] = s;
    }
}

// ---------------------------------------------------------------------------
// Kernel 2: fold partials into per-channel means. 1 block, 2 waves,
// wave == channel.
// ---------------------------------------------------------------------------
__global__ void dc_final_kernel(const float* __restrict__ partial,
                                float* __restrict__ means,
                                int P, float invCount) {
    const int lane = threadIdx.x & 31;
    const int ch   = threadIdx.x >> 5;
    float s = 0.0f;
    for (int i = lane; i < P; i += 32) s += partial[ch * P + i];
    for (int off = 16; off > 0; off >>= 1)
        s += __shfl_xor(s, off, 32);
    if (lane == 0) means[ch] = s * invCount;
}

// ---------------------------------------------------------------------------
// Kernel 3: streaming transform. float4 (b128) NT loads/stores; out is the
// flat concatenation [y_out | c_out]. DC slots in c get the contrast blend.
// ---------------------------------------------------------------------------
__global__ void transform_kernel(const float* __restrict__ y,
                                 const float* __restrict__ c,
                                 float* __restrict__ out,
                                 const float* __restrict__ means,
                                 long long yN, long long cN, long long totalVec) {
    const float F = 1.9f;
    const long long chanElems = cN >> 1;
    const long long stride = (long long)gridDim.x * blockDim.x;

    for (long long v = (long long)blockIdx.x * blockDim.x + threadIdx.x;
         v < totalVec; v += stride) {
        const long long e = v << 2;
        if (e < yN) {
            v4f x = __builtin_nontemporal_load((const v4f*)y + v);
            v4f r = clamp4(clamp4(x) * F);
            __builtin_nontemporal_store(r, (v4f*)out + v);
        } else {
            const long long ec = e - yN;
            v4f x = __builtin_nontemporal_load((const v4f*)c + (ec >> 2));
            x = clamp4(x);
            v4f r = x * F;
            if ((ec & 63) == 0) {                 // DC coefficient of 8x8 block
                const int ch = (ec >= chanElems) ? 1 : 0;
                r.x = x.x * F - 0.9f * means[ch]; // (1-F)*dc_mean, F=1.9
            }
            r = clamp4(r);
            __builtin_nontemporal_store(r, (v4f*)out + v);
        }
    }
}

extern "C" void kernel_launch(void* const* d_in, const int* in_sizes, int n_in,
                              void* d_out, int out_size, void* d_ws, size_t ws_size,
                              hipStream_t stream) {
    const float* y = (const float*)d_in[0];
    const float* c = (const float*)d_in[1];
    float* out = (float*)d_out;

    const long long yN = in_sizes[0];        // 1*512*512*64 = 16,777,216
    const long long cN = in_sizes[1];        // 2*256*256*64 =  8,388,608
    const long long chanElems = cN / 2;      // 4,194,304
    const int dcPerChannel = (int)(chanElems / 64);   // 65,536

    float* partial = (float*)d_ws;           // [2 * G] partial sums
    float* means   = partial + 128;          // [2] per-channel DC means

    const int G = 64;                        // reduction groups per channel
    const int dcPerBlock = dcPerChannel / G; // 1024 DC values per block

    dim3 g1(G, 2);
    dc_partial_kernel<<<g1, 256, 0, stream>>>(c, partial, dcPerBlock, chanElems);
    dc_final_kernel<<<1, 64, 0, stream>>>(partial, means, G, 1.0f / (float)dcPerChannel);

    const long long totalVec = (yN + cN) >> 2;   // 6,291,456 float4s
    transform_kernel<<<6144, 256, 0, stream>>>(y, c, out, means, yN, cN, totalVec);
}